// SGC_74869869904020
// MI455X (gfx1250) — compile-verified
//
#include <hip/hip_runtime.h>
#include <hip/hip_bf16.h>

#define N_NODES 50000
#define D_FEAT  128
#define C_OUT   40
#define C_PAD   48        // 3 full 16-col WMMA tiles, zero-padded
#define WT_STRIDE 388     // dwords; 388 % 64 == 4 -> conflict-free b64 lanes

typedef __attribute__((ext_vector_type(2))) float v2f;
typedef __attribute__((ext_vector_type(8))) float v8f;

// ---------------------------------------------------------------------------
// Algebraic restructuring: out = sum_k A_k (x W_k) + b.
// Stage 1 computes y_k = x @ W_k  ([N,40] per k) with f32 WMMA.
// One wave per 16-row strip of x, all 3 (padded) col-tiles per wave.
// W_k is staged TRANSPOSED in LDS (Wt[col][k], stride 388 dwords) so each
// B fragment {K,K+1} is one aligned ds_load_b64 into an even VGPR pair:
// no repacking movs, and the stride-mod-64==4 padding makes all 32 lanes hit
// distinct bank pairs (conflict-free across both K-halves).
//
// f32 16x16x4 fragment layouts (wave32):
//   A 16x4 : lanes 0-15 -> M=lane, {K0,K1}; lanes 16-31 -> M=lane-16, {K2,K3}
//   B 4x16 : lanes 0-15 -> N=lane, {K0,K1}; lanes 16-31 -> N=lane-16, {K2,K3}
//   C/D    : VGPR r, lanes 0-15 -> (M=r, N=lane); lanes 16-31 -> (M=r+8, N=lane-16)
// ---------------------------------------------------------------------------
__global__ void sgc_xw_wmma(const float* __restrict__ x,
                            const float* __restrict__ W,   // [3*128, 40]
                            float*       __restrict__ y,   // [3][N, 40]
                            int nTilesM) {
    __shared__ float Wt[C_PAD * WT_STRIDE];   // 48*388*4 = 74496 B (< 320KB WGP)

    int plane = blockIdx.y;                   // which W_k slice / y plane
    const float* Wk = W + (long)plane * D_FEAT * C_OUT;

    for (int i = threadIdx.x; i < C_PAD * D_FEAT; i += blockDim.x) {
        int cc = i >> 7;                      // / D_FEAT
        int kk = i & (D_FEAT - 1);
        Wt[cc * WT_STRIDE + kk] = (cc < C_OUT) ? Wk[kk * C_OUT + cc] : 0.0f;
    }
    __syncthreads();

    int tM = blockIdx.x * (blockDim.x >> 5) + (threadIdx.x >> 5);
    if (tM >= nTilesM) return;       // whole-wave predicate: EXEC all-1s inside

    int lane = threadIdx.x & 31;
    int half = lane >> 4;            // 0 -> K pair {0,1}, 1 -> K pair {2,3}
    int l16  = lane & 15;

    // N_NODES % 16 == 0: every row valid, no guard.
    const float* xrow = x + (long)(tM * 16 + l16) * D_FEAT + half * 2;
    const float* b0p  = &Wt[(l16)      * WT_STRIDE + half * 2];
    const float* b1p  = &Wt[(l16 + 16) * WT_STRIDE + half * 2];
    const float* b2p  = &Wt[(l16 + 32) * WT_STRIDE + half * 2];

    v8f acc0 = {}, acc1 = {}, acc2 = {};
    #pragma unroll 4
    for (int k = 0; k < D_FEAT; k += 4) {
        v2f a  = *(const v2f*)(xrow + k);     // global_load_b64
        v2f b0 = *(const v2f*)(b0p + k);      // ds_load_b64, conflict-free
        v2f b1 = *(const v2f*)(b1p + k);
        v2f b2 = *(const v2f*)(b2p + k);
        acc0 = __builtin_amdgcn_wmma_f32_16x16x4_f32(false, a, false, b0,
                                                     (short)0, acc0, false, false);
        acc1 = __builtin_amdgcn_wmma_f32_16x16x4_f32(false, a, false, b1,
                                                     (short)0, acc1, false, false);
        acc2 = __builtin_amdgcn_wmma_f32_16x16x4_f32(false, a, false, b2,
                                                     (short)0, acc2, false, false);
    }

    float* yp = y + (long)plane * N_NODES * C_OUT;
    int baseM = tM * 16 + half * 8;
    #pragma unroll
    for (int r = 0; r < 8; ++r) {
        float* orow = yp + (long)(baseM + r) * C_OUT;
        orow[l16]      = acc0[r];
        orow[l16 + 16] = acc1[r];
        if (l16 + 32 < C_OUT)
            orow[l16 + 32] = acc2[r];
    }
}

// ---------------------------------------------------------------------------
// out[n, f] = bias[f]  (accumulation base; also handles rows with no edges)
// ---------------------------------------------------------------------------
__global__ void sgc_init_out(float* __restrict__ out,
                             const float* __restrict__ bias, unsigned total) {
    unsigned i = blockIdx.x * blockDim.x + threadIdx.x;
    if (i >= total) return;
    unsigned f = i % C_OUT;          // strength-reduced constant division
    out[i] = bias[f];
}

// ---------------------------------------------------------------------------
// Scatter: out[rows[e], f] += vals[e] * y_k[cols[e], f], one thread per
// (edge, feature). y_k (8MB) and out (8MB) are L2-resident; 40 consecutive
// threads share the same edge-index cachelines.
// ---------------------------------------------------------------------------
__global__ void sgc_scatter(const int*   __restrict__ rows,
                            const int*   __restrict__ cols,
                            const float* __restrict__ vals,
                            const float* __restrict__ yk,
                            float*       __restrict__ out,
                            unsigned total) {
    unsigned tid = blockIdx.x * blockDim.x + threadIdx.x;
    if (tid >= total) return;
    unsigned edge = tid / C_OUT;
    unsigned f    = tid - edge * C_OUT;

    int   r = rows[edge];
    int   c = cols[edge];
    float v = vals[edge];
    unsafeAtomicAdd(out + (long)r * C_OUT + f,
                    v * yk[(long)c * C_OUT + f]);     // global_atomic_add_f32
}

extern "C" void kernel_launch(void* const* d_in, const int* in_sizes, int n_in,
                              void* d_out, int out_size, void* d_ws, size_t ws_size,
                              hipStream_t stream) {
    const float* x  = (const float*)d_in[0];
    const int*   rows[3] = {(const int*)d_in[1], (const int*)d_in[4], (const int*)d_in[7]};
    const int*   cols[3] = {(const int*)d_in[2], (const int*)d_in[5], (const int*)d_in[8]};
    const float* vals[3] = {(const float*)d_in[3], (const float*)d_in[6], (const float*)d_in[9]};
    const float* W  = (const float*)d_in[10];
    const float* b  = (const float*)d_in[11];
    float*       out = (float*)d_out;
    float*       y   = (float*)d_ws;    // [3][N,40] f32 = 24 MB scratch

    int threads = 256;                  // 8 waves/block (wave32)
    int wavesPerBlock = threads / 32;

    // 1) y_k = x @ W_k for k=0..2 (WMMA), grid.y = plane
    int nTilesM = N_NODES / 16;         // 3125, exact
    dim3 gGrid((nTilesM + wavesPerBlock - 1) / wavesPerBlock, 3);
    sgc_xw_wmma<<<gGrid, threads, 0, stream>>>(x, W, y, nTilesM);

    // 2) out = broadcast bias
    unsigned outElems = (unsigned)N_NODES * C_OUT;
    sgc_init_out<<<(outElems + threads - 1) / threads, threads, 0, stream>>>(
        out, b, outElems);

    // 3) out[r] += v * y_k[c] for every edge (f32 atomics into L2)
    int E = in_sizes[1];
    unsigned total = (unsigned)E * C_OUT;
    unsigned sBlocks = (total + threads - 1) / threads;
    for (int k = 0; k < 3; ++k)
        sgc_scatter<<<sBlocks, threads, 0, stream>>>(
            rows[k], cols[k], vals[k],
            y + (long)k * N_NODES * C_OUT, out, total);
}